// STAR_65927747994226
// MI455X (gfx1250) — compile-verified
//
#include <hip/hip_runtime.h>
#include <hip/hip_bf16.h>

// ---------------------------------------------------------------------------
// STAR/SMPL forward pass for gfx1250 (CDNA5, wave32, WMMA).
// Heavy GEMMs (shapedirs*betas, posedirs*pose_feat) use V_WMMA_F32_16X16X4_F32
// with fully padded operands: no guards, no EXEC divergence, full unroll.
// ---------------------------------------------------------------------------

typedef float v2f __attribute__((ext_vector_type(2)));
typedef float v8f __attribute__((ext_vector_type(8)));

#define BATCH 512
#define NV    6890
#define NJNT  24
#define MROWS (NV * 3)               // 20670
#define MPAD  20672                  // MROWS padded to multiple of 16
#define OUTSTRIDE ((NV + NJNT) * 3)  // 6914*3 = 20742

static constexpr int PAR[23] = {0, 0, 0, 1, 2, 3, 4, 5, 6, 7, 8,
                                9, 9, 9, 12, 13, 14, 16, 17, 18, 19, 20, 21};

// ---------------------------------------------------------------------------
// Zero-padded copy: dst (Mpad x ldpad) <- src (Msrc x ldsrc), zeros elsewhere.
// ---------------------------------------------------------------------------
__global__ void pad_copy_kernel(const float* __restrict__ src, float* __restrict__ dst,
                                int Msrc, int ldsrc, int Mpad, int ldpad) {
  const int idx = blockIdx.x * blockDim.x + threadIdx.x;
  if (idx >= Mpad * ldpad) return;
  const int row = idx / ldpad, c = idx - row * ldpad;
  dst[idx] = (row < Msrc && c < ldsrc) ? src[row * ldsrc + c] : 0.f;
}

// ---------------------------------------------------------------------------
// WMMA f32 GEMM: D[p][b] = sum_k A[p][k] * Bt[b][k]  (+ bias)
//   A:  row-major MPAD x LDA, zero-padded (LDA = 4*KSTEPS)
//   Bt: row-major 512 x LDB  (B^T), zero-padded
//   D:  row-major MPAD x 512
// BIAS: 0 none; 1 per-row bias[row] (guarded read); 2 per-element bias[row*512+col]
// One wave = one 16x16 tile. K stepped by 4 with V_WMMA_F32_16X16X4_F32.
// A 16x4 frag: lane&15 = M row; lanes>=16 hold K=+2/+3.  (2 VGPRs)
// B 4x16 frag: lane&15 = N col; lanes>=16 hold K=+2/+3.  (2 VGPRs)
// D 16x16: VGPR g -> row M0+g (lanes 0-15) / M0+8+g (lanes 16-31), col N0+(lane&15).
// ---------------------------------------------------------------------------
template <int KSTEPS, int LDA, int LDB, int BIAS>
__global__ __launch_bounds__(256)
void wmma_gemm_kernel(const float* __restrict__ A, const float* __restrict__ Bt,
                      const float* bias, float* D) {
  const int lane = threadIdx.x & 31;
  const int wave = threadIdx.x >> 5;
  const int N0 = blockIdx.x * 16;
  const int mtile = blockIdx.y * 8 + wave;
  const int M0 = mtile * 16;
  if (M0 >= MPAD) return;  // wave-uniform exit

  const int r   = lane & 15;
  const int hi  = lane >> 4;  // 0 for lanes 0-15, 1 for 16-31
  const int col = N0 + r;

  const float* ap = A + (size_t)(M0 + r) * LDA + 2 * hi;
  const float* bp = Bt + (size_t)col * LDB + 2 * hi;

  v8f acc = {0.f, 0.f, 0.f, 0.f, 0.f, 0.f, 0.f, 0.f};

#pragma unroll
  for (int s = 0; s < KSTEPS; ++s) {
    const v2f a = *(const v2f*)(ap + s * 4);
    const v2f b = *(const v2f*)(bp + s * 4);
    acc = __builtin_amdgcn_wmma_f32_16x16x4_f32(
        /*neg_a=*/false, a, /*neg_b=*/false, b,
        /*c_mod=*/(short)0, acc, /*reuse_a=*/false, /*reuse_b=*/false);
  }

#pragma unroll
  for (int g = 0; g < 8; ++g) {
    const int row = M0 + g + 8 * hi;
    float v = acc[g];
    if (BIAS == 1) v += (row < MROWS) ? bias[row] : 0.f;  // cndmask, no branch
    if (BIAS == 2) v += bias[(size_t)row * BATCH + col];
    D[(size_t)row * BATCH + col] = v;
  }
}

// ---------------------------------------------------------------------------
// J[b][j][k] = sum_v Jreg[j][v] * vsh[(v*3+k)*512 + b]
// grid: (NJ, B), block 256, LDS tree reduce.
// ---------------------------------------------------------------------------
__global__ __launch_bounds__(256)
void jreg_kernel(const float* __restrict__ Jreg, const float* __restrict__ vsh,
                 float* __restrict__ Jb) {
  const int j = blockIdx.x, b = blockIdx.y, tid = threadIdx.x;
  float s0 = 0.f, s1 = 0.f, s2 = 0.f;
  for (int v = tid; v < NV; v += 256) {
    const float w = Jreg[j * NV + v];
    const int base = v * 3 * BATCH + b;
    s0 += w * vsh[base];
    s1 += w * vsh[base + BATCH];
    s2 += w * vsh[base + 2 * BATCH];
  }
  __shared__ float red[3 * 256];
  red[tid] = s0; red[256 + tid] = s1; red[512 + tid] = s2;
  __syncthreads();
  for (int off = 128; off > 0; off >>= 1) {
    if (tid < off) {
      red[tid]       += red[tid + off];
      red[256 + tid] += red[256 + tid + off];
      red[512 + tid] += red[512 + tid + off];
    }
    __syncthreads();
  }
  if (tid == 0) {
    float* o = Jb + b * (NJNT * 3) + j * 3;
    o[0] = red[0]; o[1] = red[256]; o[2] = red[512];
  }
}

// ---------------------------------------------------------------------------
// Rodrigues + quaternion pose features. Thread per (b, joint).
// Rbuf: [b][24][9];  pf: [b][96] (93 valid: qf[1..23][0..3] then betas[b][1])
// ---------------------------------------------------------------------------
__global__ void rodrigues_kernel(const float* __restrict__ pose,
                                 const float* __restrict__ rot,
                                 const float* __restrict__ betas,
                                 float* __restrict__ Rbuf,
                                 float* __restrict__ pf) {
  const int t = blockIdx.x * blockDim.x + threadIdx.x;
  if (t >= BATCH * NJNT) return;
  const int b = t / NJNT, j = t % NJNT;

  float tx, ty, tz;
  if (j == 0) {
    tx = rot[b * 3]; ty = rot[b * 3 + 1]; tz = rot[b * 3 + 2];
  } else {
    const int o = b * 69 + 3 * (j - 1);
    tx = pose[o]; ty = pose[o + 1]; tz = pose[o + 2];
  }
  const float ax = tx + 1e-8f, ay = ty + 1e-8f, az = tz + 1e-8f;
  const float angle = sqrtf(ax * ax + ay * ay + az * az);
  const float inv = 1.0f / angle;
  const float nx = tx * inv, ny = ty * inv, nz = tz * inv;
  const float half = 0.5f * angle;
  const float c = cosf(half), s = sinf(half);
  const float sx = s * nx, sy = s * ny, sz = s * nz;

  // quat feature (s*n, c-1): flattened per batch as (24*4); indices 4.. used
  if (j >= 1) {
    float* q = pf + b * 96 + (j * 4 - 4);
    q[0] = sx; q[1] = sy; q[2] = sz; q[3] = c - 1.0f;
  } else {
    float* q = pf + b * 96;
    q[92] = betas[b * 10 + 1];
    q[93] = 0.f; q[94] = 0.f; q[95] = 0.f;
  }

  // normalized quaternion -> rotation matrix
  const float qn = sqrtf(c * c + sx * sx + sy * sy + sz * sz);
  const float iq = 1.0f / qn;
  const float w = c * iq, x = sx * iq, y = sy * iq, z = sz * iq;
  float* R = Rbuf + (b * NJNT + j) * 9;
  R[0] = w * w + x * x - y * y - z * z;
  R[1] = 2.f * (x * y - w * z);
  R[2] = 2.f * (w * y + x * z);
  R[3] = 2.f * (w * z + x * y);
  R[4] = w * w - x * x + y * y - z * z;
  R[5] = 2.f * (y * z - w * x);
  R[6] = 2.f * (x * z - w * y);
  R[7] = 2.f * (w * x + y * z);
  R[8] = w * w - x * x - y * y + z * z;
}

// ---------------------------------------------------------------------------
// Kinematic chain: thread per batch. Fully unrolled 24-joint chain with
// compile-time parent indices. Emits G_skin (3x4/joint) and posed joints.
// ---------------------------------------------------------------------------
__global__ __launch_bounds__(256)
void chain_kernel(const float* __restrict__ Jb, const float* __restrict__ Rbuf,
                  float* __restrict__ Gsk, float* __restrict__ PJ) {
  const int b = blockIdx.x * blockDim.x + threadIdx.x;
  if (b >= BATCH) return;

  float J[NJNT][3], Jrel[NJNT][3];
#pragma unroll
  for (int j = 0; j < NJNT; ++j)
    for (int k = 0; k < 3; ++k) J[j][k] = Jb[b * (NJNT * 3) + j * 3 + k];
  Jrel[0][0] = J[0][0]; Jrel[0][1] = J[0][1]; Jrel[0][2] = J[0][2];
#pragma unroll
  for (int j = 1; j < NJNT; ++j)
    for (int k = 0; k < 3; ++k) Jrel[j][k] = J[j][k] - J[PAR[j - 1]][k];

  const float* Rb = Rbuf + b * NJNT * 9;
  float G[NJNT][12];  // rows 0..2 of each global 4x4 (bottom row is 0001)

#pragma unroll
  for (int p = 0; p < 3; ++p) {
    G[0][p * 4 + 0] = Rb[p * 3 + 0];
    G[0][p * 4 + 1] = Rb[p * 3 + 1];
    G[0][p * 4 + 2] = Rb[p * 3 + 2];
    G[0][p * 4 + 3] = Jrel[0][p];
  }
#pragma unroll
  for (int i = 1; i < NJNT; ++i) {
    const int par = PAR[i - 1];
    const float* Rl = Rb + i * 9;
#pragma unroll
    for (int p = 0; p < 3; ++p) {
      const float g0 = G[par][p * 4 + 0], g1 = G[par][p * 4 + 1],
                  g2 = G[par][p * 4 + 2], g3 = G[par][p * 4 + 3];
#pragma unroll
      for (int q = 0; q < 3; ++q)
        G[i][p * 4 + q] = g0 * Rl[0 * 3 + q] + g1 * Rl[1 * 3 + q] + g2 * Rl[2 * 3 + q];
      G[i][p * 4 + 3] = g0 * Jrel[i][0] + g1 * Jrel[i][1] + g2 * Jrel[i][2] + g3;
    }
  }

#pragma unroll
  for (int j = 0; j < NJNT; ++j) {
    float* gs = Gsk + b * (NJNT * 12) + j * 12;
    float* pj = PJ + b * (NJNT * 3) + j * 3;
#pragma unroll
    for (int p = 0; p < 3; ++p) {
      const float t = G[j][p * 4 + 0] * J[j][0] + G[j][p * 4 + 1] * J[j][1] +
                      G[j][p * 4 + 2] * J[j][2];
      pj[p] = G[j][p * 4 + 3];
      gs[p * 4 + 0] = G[j][p * 4 + 0];
      gs[p * 4 + 1] = G[j][p * 4 + 1];
      gs[p * 4 + 2] = G[j][p * 4 + 2];
      gs[p * 4 + 3] = G[j][p * 4 + 3] - t;
    }
  }
}

// ---------------------------------------------------------------------------
// LBS skinning: thread per (b, v). grid(ceil(V/256), B).
// T = sum_j weights[v][j] * G_skin[b][j]  (3x4);  out = T * [v_posed,1] + tr.
// Coalesced output stores (lanes over v).
// ---------------------------------------------------------------------------
__global__ __launch_bounds__(256)
void skin_kernel(const float* __restrict__ vposed, const float* __restrict__ wts,
                 const float* __restrict__ Gsk, const float* __restrict__ trans,
                 float* __restrict__ out) {
  const int v = blockIdx.x * blockDim.x + threadIdx.x;
  const int b = blockIdx.y;
  if (v >= NV) return;

  float T[12];
#pragma unroll
  for (int i = 0; i < 12; ++i) T[i] = 0.f;
  const float* Gb = Gsk + b * (NJNT * 12);
  for (int j = 0; j < NJNT; ++j) {
    const float wj = wts[v * NJNT + j];
    const float* g = Gb + j * 12;
#pragma unroll
    for (int i = 0; i < 12; ++i) T[i] += wj * g[i];
  }
  const int base = v * 3 * BATCH + b;
  const float x = vposed[base];
  const float y = vposed[base + BATCH];
  const float z = vposed[base + 2 * BATCH];
  float* o = out + b * OUTSTRIDE + v * 3;
#pragma unroll
  for (int p = 0; p < 3; ++p)
    o[p] = T[p * 4] * x + T[p * 4 + 1] * y + T[p * 4 + 2] * z + T[p * 4 + 3] +
           trans[b * 3 + p];
}

// ---------------------------------------------------------------------------
// Joint output with fixups: thread per (b, j).
// Jt[12] = mean(Jt[12:15]); Jt[0] = mean(Jt[0:3]); Jt[15] = skinned v[331].
// ---------------------------------------------------------------------------
__global__ void joints_kernel(const float* __restrict__ PJ,
                              const float* __restrict__ trans,
                              float* __restrict__ out) {
  const int t = blockIdx.x * blockDim.x + threadIdx.x;
  if (t >= BATCH * NJNT) return;
  const int b = t / NJNT, j = t % NJNT;
  const float* pj = PJ + b * (NJNT * 3);
  float* o = out + b * OUTSTRIDE + (NV + j) * 3;
#pragma unroll
  for (int k = 0; k < 3; ++k) {
    float val;
    const float tr = trans[b * 3 + k];
    if (j == 12) {
      val = (pj[12 * 3 + k] + pj[13 * 3 + k] + pj[14 * 3 + k]) * (1.f / 3.f) + tr;
    } else if (j == 0) {
      val = (pj[0 * 3 + k] + pj[1 * 3 + k] + pj[2 * 3 + k]) * (1.f / 3.f) + tr;
    } else if (j == 15) {
      val = out[b * OUTSTRIDE + 331 * 3 + k];  // already includes translation
    } else {
      val = pj[j * 3 + k] + tr;
    }
    o[k] = val;
  }
}

// ---------------------------------------------------------------------------
extern "C" void kernel_launch(void* const* d_in, const int* in_sizes, int n_in,
                              void* d_out, int out_size, void* d_ws, size_t ws_size,
                              hipStream_t stream) {
  const float* pose  = (const float*)d_in[0];  // (512,69)
  const float* betas = (const float*)d_in[1];  // (512,10)
  const float* rot   = (const float*)d_in[2];  // (512,3)
  const float* trans = (const float*)d_in[3];  // (512,3)
  const float* vt    = (const float*)d_in[4];  // (6890,3)
  const float* shd   = (const float*)d_in[5];  // (6890,3,10)
  const float* psd   = (const float*)d_in[6];  // (20670,93)
  const float* jrg   = (const float*)d_in[7];  // (24,6890)
  const float* wts   = (const float*)d_in[8];  // (6890,24)
  float* out = (float*)d_out;

  float* w = (float*)d_ws;
  float* vsh  = w;                               // MPAD*512 (v_shaped -> v_posed in place)
  float* pf   = vsh + (size_t)MPAD * BATCH;      // 512*96 pose features (zero-padded)
  float* Rb   = pf + BATCH * 96;                 // 512*24*9
  float* Jb   = Rb + BATCH * NJNT * 9;           // 512*24*3
  float* Gsk  = Jb + BATCH * NJNT * 3;           // 512*24*12
  float* PJ   = Gsk + BATCH * NJNT * 12;         // 512*24*3
  float* A1p  = PJ + BATCH * NJNT * 3;           // MPAD*12  padded shapedirs
  float* A2p  = A1p + (size_t)MPAD * 12;         // MPAD*96  padded posedirs
  float* Bbp  = A2p + (size_t)MPAD * 96;         // 512*12   padded betas

  const int mtiles = MPAD / 16;                  // 1292
  dim3 gemm_grid(BATCH / 16, (mtiles + 7) / 8);  // (32, 162)

  // 0. zero-padded operand copies (unconditional GEMM inner loops)
  pad_copy_kernel<<<((MPAD * 12) + 255) / 256, 256, 0, stream>>>(shd, A1p, MROWS, 10, MPAD, 12);
  pad_copy_kernel<<<((MPAD * 96) + 255) / 256, 256, 0, stream>>>(psd, A2p, MROWS, 93, MPAD, 96);
  pad_copy_kernel<<<((BATCH * 12) + 255) / 256, 256, 0, stream>>>(betas, Bbp, BATCH, 10, BATCH, 12);

  // 1. v_shaped = shapedirs * betas^T + v_template   (WMMA, K=12, 3 steps)
  wmma_gemm_kernel<3, 12, 12, 1><<<gemm_grid, 256, 0, stream>>>(A1p, Bbp, vt, vsh);
  // 2. J = J_regressor * v_shaped
  jreg_kernel<<<dim3(NJNT, BATCH), 256, 0, stream>>>(jrg, vsh, Jb);
  // 3. Rodrigues rotations + pose features
  rodrigues_kernel<<<(BATCH * NJNT + 255) / 256, 256, 0, stream>>>(pose, rot, betas, Rb, pf);
  // 4. Kinematic chain -> G_skin, posed joints
  chain_kernel<<<(BATCH + 255) / 256, 256, 0, stream>>>(Jb, Rb, Gsk, PJ);
  // 5. v_posed = v_shaped + posedirs * pose_feat^T   (WMMA, K=96, 24 steps, in place)
  wmma_gemm_kernel<24, 96, 96, 2><<<gemm_grid, 256, 0, stream>>>(A2p, pf, vsh, vsh);
  // 6. LBS skinning -> out vertices
  skin_kernel<<<dim3((NV + 255) / 256, BATCH), 256, 0, stream>>>(vsh, wts, Gsk, trans, out);
  // 7. joints with fixups
  joints_kernel<<<(BATCH * NJNT + 255) / 256, 256, 0, stream>>>(PJ, trans, out);
}